// SparseAttention_188978561463
// MI455X (gfx1250) — compile-verified
//
#include <hip/hip_runtime.h>
#include <hip/hip_bf16.h>

// ---------------- problem constants ----------------
#define BATCH     2
#define SEQ       2048
#define DIM       1024
#define N_HEADS   16
#define HEAD_DIM  64
#define K_KEEP    409                    // max(1, int(2048*0.2))
#define SCALE_F   0.125f                 // 64^-0.5

typedef __attribute__((ext_vector_type(2))) float v2f;
typedef __attribute__((ext_vector_type(8))) float v8f;

// D = A(16x4, f32) * B(4x16, f32) + C(16x16, f32), wave32.
// A layout: lane = (k>=2)*16 + m ; a.x = A[m][2*kh], a.y = A[m][2*kh+1]
// B layout: lane = (k>=2)*16 + n ; b.x = B[2*kh][n], b.y = B[2*kh+1][n]
// C layout: lane = mh*16 + n ; c[r] = C[r + 8*mh][n]
__device__ __forceinline__ v8f wmma_f32(v2f a, v2f b, v8f c) {
  return __builtin_amdgcn_wmma_f32_16x16x4_f32(
      /*neg_a=*/false, a, /*neg_b=*/false, b,
      /*c_mod=*/(short)0, c, /*reuse_a=*/false, /*reuse_b=*/false);
}

__device__ __forceinline__ v2f ld2(const float* __restrict__ p) {
  v2f r; r.x = p[0]; r.y = p[1]; return r;
}

// ---------------------------------------------------------------------------
// K1: Q/K/V projections.  y[m][e] = sum_d x[m][d] * w[e][d] + bias[e]
// 2x2 register-blocked: one wave -> 32x32 output tile (4 WMMA / 4 frag loads).
// Output layout: [B, H, S, hd].  grid = (BS/32, DIM/32, 3), block = 32.
// ---------------------------------------------------------------------------
__global__ __launch_bounds__(32) void qkv_proj(
    const float* __restrict__ x,
    const float* __restrict__ wq, const float* __restrict__ bq,
    const float* __restrict__ wk, const float* __restrict__ bk,
    const float* __restrict__ wv, const float* __restrict__ bv,
    float* __restrict__ Q, float* __restrict__ K, float* __restrict__ V) {
  const int lane = threadIdx.x;
  const int m0 = blockIdx.x * 32;       // row tile over B*S
  const int e0 = blockIdx.y * 32;       // col tile over DIM
  const int which = blockIdx.z;         // 0=Q 1=K 2=V

  const float* w    = (which == 0) ? wq : (which == 1) ? wk : wv;
  const float* bias = (which == 0) ? bq : (which == 1) ? bk : bv;
  float*       dst  = (which == 0) ? Q  : (which == 1) ? K  : V;

  const int n  = lane & 15;
  const int kh = lane >> 4;

  const float* arow0 = x + (size_t)(m0 + n) * DIM;
  const float* arow1 = arow0 + (size_t)16 * DIM;
  const float* brow0 = w + (size_t)(e0 + n) * DIM;
  const float* brow1 = brow0 + (size_t)16 * DIM;

  v8f c00 = {}, c01 = {}, c10 = {}, c11 = {};
  for (int k0 = 0; k0 < DIM; k0 += 4) {
    const int ka = k0 + 2 * kh;
    const v2f a0 = ld2(arow0 + ka);
    const v2f a1 = ld2(arow1 + ka);
    const v2f b0 = ld2(brow0 + ka);
    const v2f b1 = ld2(brow1 + ka);
    c00 = wmma_f32(a0, b0, c00);
    c01 = wmma_f32(a0, b1, c01);
    c10 = wmma_f32(a1, b0, c10);
    c11 = wmma_f32(a1, b1, c11);
  }

#pragma unroll
  for (int tj = 0; tj < 2; ++tj) {
    const int e = e0 + tj * 16 + n;
    const float be = bias[e];
    const int h = e >> 6, dd = e & 63;
#pragma unroll
    for (int ti = 0; ti < 2; ++ti) {
      const v8f& c = (ti == 0) ? (tj == 0 ? c00 : c01) : (tj == 0 ? c10 : c11);
#pragma unroll
      for (int r = 0; r < 8; ++r) {
        const int m = m0 + ti * 16 + r + 8 * kh;
        const int b_ = m >> 11, s = m & 2047;           // m = b*S + s
        dst[(((size_t)(b_ * N_HEADS + h) * SEQ + s) * HEAD_DIM) + dd] = c[r] + be;
      }
    }
  }
}

// ---------------------------------------------------------------------------
// K2: scores = Q K^T * scale, per (b,h).  2x2 blocked -> 32x32 score tile.
// grid = (S/32, S/32, B*H), block = 32.
// ---------------------------------------------------------------------------
__global__ __launch_bounds__(32) void qk_scores(
    const float* __restrict__ Q, const float* __restrict__ K,
    float* __restrict__ scores) {
  const int lane = threadIdx.x;
  const int q0 = blockIdx.x * 32;
  const int c0 = blockIdx.y * 32;
  const int bh = blockIdx.z;
  const int n  = lane & 15;
  const int kh = lane >> 4;

  const float* qrow0 = Q + ((size_t)bh * SEQ + (q0 + n)) * HEAD_DIM;
  const float* qrow1 = qrow0 + (size_t)16 * HEAD_DIM;
  const float* krow0 = K + ((size_t)bh * SEQ + (c0 + n)) * HEAD_DIM;
  const float* krow1 = krow0 + (size_t)16 * HEAD_DIM;

  v8f c00 = {}, c01 = {}, c10 = {}, c11 = {};
#pragma unroll
  for (int k0 = 0; k0 < HEAD_DIM; k0 += 4) {
    const int ka = k0 + 2 * kh;
    const v2f a0 = ld2(qrow0 + ka);
    const v2f a1 = ld2(qrow1 + ka);
    const v2f b0 = ld2(krow0 + ka);
    const v2f b1 = ld2(krow1 + ka);
    c00 = wmma_f32(a0, b0, c00);
    c01 = wmma_f32(a0, b1, c01);
    c10 = wmma_f32(a1, b0, c10);
    c11 = wmma_f32(a1, b1, c11);
  }

  const size_t base = (size_t)bh * SEQ * SEQ;
#pragma unroll
  for (int ti = 0; ti < 2; ++ti) {
#pragma unroll
    for (int tj = 0; tj < 2; ++tj) {
      const v8f& c = (ti == 0) ? (tj == 0 ? c00 : c01) : (tj == 0 ? c10 : c11);
#pragma unroll
      for (int r = 0; r < 8; ++r) {
        const int qm = q0 + ti * 16 + r + 8 * kh;
        scores[base + (size_t)qm * SEQ + (c0 + tj * 16 + n)] = c[r] * SCALE_F;
      }
    }
  }
}

// ---------------------------------------------------------------------------
// K3: exact top-k (k=409) per row via 32-pass radix select, then masked
// softmax in place.  One 256-thread block per row (B*H*S rows).
// ---------------------------------------------------------------------------
__device__ __forceinline__ unsigned f2key(float f) {
  unsigned b = __float_as_uint(f);
  return (b & 0x80000000u) ? ~b : (b | 0x80000000u);   // monotone mapping
}

__global__ __launch_bounds__(256) void topk_softmax(float* __restrict__ attn) {
  __shared__ float    s_val[SEQ];
  __shared__ unsigned s_key[SEQ];
  __shared__ int      s_cnt;
  __shared__ unsigned s_maxkey;
  __shared__ float    s_sum;

  const int row = blockIdx.x;
  const int tid = threadIdx.x;
  float* rowp = attn + (size_t)row * SEQ;

  if (tid == 0) { s_maxkey = 0u; s_sum = 0.0f; }
  unsigned lmax = 0u;
  for (int i = tid; i < SEQ; i += 256) {
    const float v = rowp[i];
    const unsigned k = f2key(v);
    s_val[i] = v;
    s_key[i] = k;
    lmax = (k > lmax) ? k : lmax;
  }
  __syncthreads();
  atomicMax(&s_maxkey, lmax);

  // radix select: exact key of the K_KEEP-th largest element
  unsigned prefix = 0u;
  int kk = K_KEEP;
  for (int bit = 31; bit >= 0; --bit) {
    __syncthreads();
    if (tid == 0) s_cnt = 0;
    __syncthreads();
    const unsigned want = (prefix >> bit) | 1u;   // ((prefix | 1<<bit) >> bit)
    int local = 0;
    for (int i = tid; i < SEQ; i += 256)
      local += ((s_key[i] >> bit) == want) ? 1 : 0;
    atomicAdd(&s_cnt, local);
    __syncthreads();
    const int c = s_cnt;
    if (kk <= c) prefix |= (1u << bit);
    else         kk -= c;
  }
  const unsigned thresh = prefix;

  // stable masked softmax: row max is top-1, always selected
  const unsigned mk = s_maxkey;
  const unsigned mb = (mk & 0x80000000u) ? (mk & 0x7fffffffu) : ~mk;
  const float rmax = __uint_as_float(mb);

  float lsum = 0.0f;
  for (int i = tid; i < SEQ; i += 256)
    if (s_key[i] >= thresh) lsum += __expf(s_val[i] - rmax);
  atomicAdd(&s_sum, lsum);
  __syncthreads();
  const float inv = 1.0f / s_sum;

  for (int i = tid; i < SEQ; i += 256)
    rowp[i] = (s_key[i] >= thresh) ? __expf(s_val[i] - rmax) * inv : 0.0f;
}

// ---------------------------------------------------------------------------
// K4: ctx = attn @ V per (b,h).  One wave -> 32 rows x all 64 cols.
// 8 accumulators; the 4 B-fragments are reused across both row tiles, and
// the streaming attn rows are prefetched (global_prefetch_b8).
// grid = (S/32, B*H), block = 32.
// ---------------------------------------------------------------------------
__global__ __launch_bounds__(32) void attn_v(
    const float* __restrict__ attn, const float* __restrict__ V,
    float* __restrict__ ctx) {
  const int lane = threadIdx.x;
  const int q0 = blockIdx.x * 32;
  const int bh = blockIdx.y;
  const int n  = lane & 15;
  const int kh = lane >> 4;

  const float* arow0 = attn + ((size_t)bh * SEQ + (q0 + n)) * SEQ;
  const float* arow1 = arow0 + (size_t)16 * SEQ;
  const float* vbase = V + (size_t)bh * SEQ * HEAD_DIM;

  v8f a0c0 = {}, a0c1 = {}, a0c2 = {}, a0c3 = {};
  v8f a1c0 = {}, a1c1 = {}, a1c2 = {}, a1c3 = {};

  for (int k0 = 0; k0 < SEQ; k0 += 4) {
    const int ka = k0 + 2 * kh;
    // prefetch the streaming operand one cacheline-ish ahead
    __builtin_prefetch(arow0 + ka + 64, 0, 1);
    __builtin_prefetch(arow1 + ka + 64, 0, 1);

    const v2f a0 = ld2(arow0 + ka);
    const v2f a1 = ld2(arow1 + ka);
    const float* v0 = vbase + (size_t)ka * HEAD_DIM;   // V row ka
    const float* v1 = v0 + HEAD_DIM;                   // V row ka+1
    v2f b;
    b.x = v0[n];      b.y = v1[n];
    a0c0 = wmma_f32(a0, b, a0c0);  a1c0 = wmma_f32(a1, b, a1c0);
    b.x = v0[n + 16]; b.y = v1[n + 16];
    a0c1 = wmma_f32(a0, b, a0c1);  a1c1 = wmma_f32(a1, b, a1c1);
    b.x = v0[n + 32]; b.y = v1[n + 32];
    a0c2 = wmma_f32(a0, b, a0c2);  a1c2 = wmma_f32(a1, b, a1c2);
    b.x = v0[n + 48]; b.y = v1[n + 48];
    a0c3 = wmma_f32(a0, b, a0c3);  a1c3 = wmma_f32(a1, b, a1c3);
  }

#pragma unroll
  for (int r = 0; r < 8; ++r) {
    const int qm0 = q0 + r + 8 * kh;
    float* o0 = ctx + ((size_t)bh * SEQ + qm0) * HEAD_DIM + n;
    o0[0]  = a0c0[r];
    o0[16] = a0c1[r];
    o0[32] = a0c2[r];
    o0[48] = a0c3[r];
    float* o1 = o0 + (size_t)16 * HEAD_DIM;
    o1[0]  = a1c0[r];
    o1[16] = a1c1[r];
    o1[32] = a1c2[r];
    o1[48] = a1c3[r];
  }
}

// ---------------------------------------------------------------------------
// K5: out = ctx(merged heads) @ Wo^T + bo.  2x2 blocked -> 32x32 tile.
// ctx is [B,H,S,hd]; logical A[m][k] with k = h*64+dd.
// grid = (BS/32, DIM/32), block = 32.
// ---------------------------------------------------------------------------
__global__ __launch_bounds__(32) void out_proj(
    const float* __restrict__ ctx, const float* __restrict__ wo,
    const float* __restrict__ bo, float* __restrict__ out) {
  const int lane = threadIdx.x;
  const int m0 = blockIdx.x * 32;
  const int e0 = blockIdx.y * 32;
  const int n  = lane & 15;
  const int kh = lane >> 4;

  const int mA = m0 + n;
  const int mB = mA + 16;
  const int bA = mA >> 11, sA = mA & 2047;
  const int bB = mB >> 11, sB = mB & 2047;
  const float* crow0 = ctx + ((size_t)bA * N_HEADS * SEQ + sA) * HEAD_DIM;
  const float* crow1 = ctx + ((size_t)bB * N_HEADS * SEQ + sB) * HEAD_DIM;
  const float* brow0 = wo + (size_t)(e0 + n) * DIM;
  const float* brow1 = brow0 + (size_t)16 * DIM;

  v8f c00 = {}, c01 = {}, c10 = {}, c11 = {};
  for (int k0 = 0; k0 < DIM; k0 += 4) {
    const int ka = k0 + 2 * kh;                // even -> ka,ka+1 share a head
    const int h = ka >> 6, dd = ka & 63;
    const size_t hoff = (size_t)h * SEQ * HEAD_DIM + dd;
    const v2f a0 = ld2(crow0 + hoff);
    const v2f a1 = ld2(crow1 + hoff);
    const v2f b0 = ld2(brow0 + ka);
    const v2f b1 = ld2(brow1 + ka);
    c00 = wmma_f32(a0, b0, c00);
    c01 = wmma_f32(a0, b1, c01);
    c10 = wmma_f32(a1, b0, c10);
    c11 = wmma_f32(a1, b1, c11);
  }

#pragma unroll
  for (int tj = 0; tj < 2; ++tj) {
    const int e = e0 + tj * 16 + n;
    const float be = bo[e];
#pragma unroll
    for (int ti = 0; ti < 2; ++ti) {
      const v8f& c = (ti == 0) ? (tj == 0 ? c00 : c01) : (tj == 0 ? c10 : c11);
#pragma unroll
      for (int r = 0; r < 8; ++r) {
        const int mm = m0 + ti * 16 + r + 8 * kh;
        out[(size_t)mm * DIM + e] = c[r] + be;
      }
    }
  }
}

// ---------------------------------------------------------------------------
extern "C" void kernel_launch(void* const* d_in, const int* in_sizes, int n_in,
                              void* d_out, int out_size, void* d_ws, size_t ws_size,
                              hipStream_t stream) {
  const float* x  = (const float*)d_in[0];
  const float* wq = (const float*)d_in[1];
  const float* bq = (const float*)d_in[2];
  const float* wk = (const float*)d_in[3];
  const float* bk = (const float*)d_in[4];
  const float* wv = (const float*)d_in[5];
  const float* bv = (const float*)d_in[6];
  const float* wo = (const float*)d_in[7];
  const float* bo = (const float*)d_in[8];

  const size_t QKV_ELEMS = (size_t)BATCH * N_HEADS * SEQ * HEAD_DIM; // 4,194,304
  float* Q   = (float*)d_ws;
  float* K   = Q + QKV_ELEMS;
  float* V   = K + QKV_ELEMS;
  float* ctx = V + QKV_ELEMS;                                        // 64 MB total

  float* out  = (float*)d_out;                                       // [B,S,DIM]
  float* attn = out + (size_t)BATCH * SEQ * DIM;                     // [B,H,S,S]

  const int BS = BATCH * SEQ;        // 4096
  const int BH = BATCH * N_HEADS;    // 32

  qkv_proj<<<dim3(BS / 32, DIM / 32, 3), 32, 0, stream>>>(
      x, wq, bq, wk, bk, wv, bv, Q, K, V);

  qk_scores<<<dim3(SEQ / 32, SEQ / 32, BH), 32, 0, stream>>>(Q, K, attn);

  topk_softmax<<<dim3(BH * SEQ), 256, 0, stream>>>(attn);

  attn_v<<<dim3(SEQ / 32, BH), 32, 0, stream>>>(attn, V, ctx);

  out_proj<<<dim3(BS / 32, DIM / 32), 32, 0, stream>>>(ctx, wo, bo, out);
}